// GCN_41188736369126
// MI455X (gfx1250) — compile-verified
//
#include <hip/hip_runtime.h>
#include <hip/hip_bf16.h>

#define D 128
#define EPSBN 1e-5f

typedef float v2f __attribute__((ext_vector_type(2)));
typedef float v8f __attribute__((ext_vector_type(8)));

// ---------------- utility kernels ----------------

__global__ void k_zero(float* __restrict__ p, long n) {
    long i = (long)blockIdx.x * blockDim.x + threadIdx.x;
    if (i < n) p[i] = 0.f;
}

__global__ void k_init_gmax(unsigned* __restrict__ g, int G) {
    int i = blockIdx.x * blockDim.x + threadIdx.x;
    if (i < G) g[i] = 0x007FFFFFu;  // order-preserving encoding of -inf
}

// ---------------- per-graph mean of initial h ----------------

__global__ void k_graph_stats(const float* __restrict__ h, const int* __restrict__ gid,
                              float* __restrict__ gsum, float* __restrict__ gcnt, int N) {
    int wid  = (int)(((long)blockIdx.x * blockDim.x + threadIdx.x) >> 5);
    int lane = threadIdx.x & 31;
    if (wid >= N) return;
    int g = gid[wid];
    float4 v = *(const float4*)(h + (size_t)wid * D + lane * 4);
    float* dst = gsum + (size_t)g * D + lane * 4;
    atomicAdd(dst + 0, v.x); atomicAdd(dst + 1, v.y);
    atomicAdd(dst + 2, v.z); atomicAdd(dst + 3, v.w);
    if (lane == 0) atomicAdd(gcnt + g, 1.f);
}

__global__ void k_init_avg(const float* __restrict__ gsum, const float* __restrict__ gcnt,
                           float* __restrict__ out, int GD) {
    int i = blockIdx.x * blockDim.x + threadIdx.x;
    if (i < GD) out[i] = gsum[i] / fmaxf(gcnt[i >> 7], 1.f);
}

// ---------------- degrees ----------------

__global__ void k_degrees(const int* __restrict__ src, const int* __restrict__ dst,
                          float* __restrict__ dout, float* __restrict__ din, int E) {
    int i = blockIdx.x * blockDim.x + threadIdx.x;
    if (i < E) {
        atomicAdd(dout + src[i], 1.f);
        atomicAdd(din + dst[i], 1.f);
    }
}

__global__ void k_deg_finalize(float* __restrict__ dout, float* __restrict__ din, int N) {
    int i = blockIdx.x * blockDim.x + threadIdx.x;
    if (i < N) {
        dout[i] = rsqrtf(fmaxf(dout[i], 1.f));
        din[i]  = rsqrtf(fmaxf(din[i], 1.f));
    }
}

// ---------------- BatchNorm (training stats over N) ----------------

__global__ void k_bn_stats(const float* __restrict__ x, float* __restrict__ bsum,
                           float* __restrict__ bsq, int N) {
    int d = threadIdx.x;  // 128 threads
    float s = 0.f, ss = 0.f;
    for (int n = blockIdx.x; n < N; n += gridDim.x) {
        float v = x[(size_t)n * D + d];
        s += v; ss += v * v;
    }
    atomicAdd(bsum + d, s);
    atomicAdd(bsq + d, ss);
}

__global__ void k_bn_finalize(const float* __restrict__ bsum, const float* __restrict__ bsq,
                              const float* __restrict__ gamma, const float* __restrict__ beta,
                              float* __restrict__ scale, float* __restrict__ shift, int N) {
    int d = threadIdx.x;
    float invN = 1.f / (float)N;
    float m = bsum[d] * invN;
    float v = fmaxf(bsq[d] * invN - m * m, 0.f);
    float sc = rsqrtf(v + EPSBN) * gamma[d];
    scale[d] = sc;
    shift[d] = beta[d] - m * sc;
}

// xs = (bn(x)) * inv_sqrt_out  (float4 elementwise)
__global__ void k_xs(const float* __restrict__ x, const float* __restrict__ scale,
                     const float* __restrict__ shift, const float* __restrict__ invout,
                     float* __restrict__ xs, int N) {
    long i = (long)blockIdx.x * blockDim.x + threadIdx.x;  // float4 index
    long total = (long)N * (D / 4);
    if (i >= total) return;
    int n  = (int)(i >> 5);
    int c4 = (int)(i & 31);
    float4 v = ((const float4*)x)[i];
    float inv = invout[n];
    const float* sc = scale + c4 * 4;
    const float* sh = shift + c4 * 4;
    float4 o;
    o.x = (v.x * sc[0] + sh[0]) * inv;
    o.y = (v.y * sc[1] + sh[1]) * inv;
    o.z = (v.z * sc[2] + sh[2]) * inv;
    o.w = (v.w * sc[3] + sh[3]) * inv;
    ((float4*)xs)[i] = o;
}

// ---------------- SpMM: agg[dst] += xs[src], one wave per edge ----------------

__global__ void k_spmm(const int* __restrict__ src, const int* __restrict__ dst,
                       const float* __restrict__ xs, float* __restrict__ agg, int E) {
    int wid  = (int)(((long)blockIdx.x * blockDim.x + threadIdx.x) >> 5);
    int lane = threadIdx.x & 31;
    if (wid >= E) return;
    int s = src[wid], d = dst[wid];
    float4 v = *(const float4*)(xs + (size_t)s * D + lane * 4);
    float* a = agg + (size_t)d * D + lane * 4;
    atomicAdd(a + 0, v.x); atomicAdd(a + 1, v.y);
    atomicAdd(a + 2, v.z); atomicAdd(a + 3, v.w);
}

// ---------------- WMMA GEMM: hout = hprev + relu((agg*inv_in) @ W + b) ----------------
// One block = 16 output rows; 8 waves, each owns one 16x16 column tile of D=128.
// K reduced with 32 chained V_WMMA_F32_16X16X4_F32 steps.

__global__ __launch_bounds__(256)
void k_gemm_relu_res(const float* __restrict__ agg, const float* __restrict__ invin,
                     const float* __restrict__ W, const float* __restrict__ bias,
                     const float* __restrict__ hprev, float* __restrict__ hout, int N) {
    __shared__ float sA[16 * 132];  // padded stride 132 -> bank-conflict-free A reads
    int row0 = blockIdx.x * 16;

    // Stage normalized A tile (Y = agg * inv_sqrt_in): 16x128 floats = 512 float4
    for (int i = threadIdx.x; i < 512; i += 256) {
        int r  = i >> 5;
        int c4 = i & 31;
        int gr = row0 + r; if (gr > N - 1) gr = N - 1;
        float4 v = *(const float4*)(agg + (size_t)gr * D + c4 * 4);
        float s = invin[gr];
        float* p = sA + r * 132 + c4 * 4;
        p[0] = v.x * s; p[1] = v.y * s; p[2] = v.z * s; p[3] = v.w * s;
    }
    __syncthreads();

    int lane = threadIdx.x & 31;
    int wave = threadIdx.x >> 5;
    int col0 = wave * 16;
    int m    = lane & 15;   // row within A fragment / col within B fragment
    int kh   = lane >> 4;   // K half-select (0: K={0,1}, 1: K={2,3})

    const float* arow = sA + m * 132 + 2 * kh;                  // A[m, k + 2*kh + {0,1}]
    const float* bcol = W + (size_t)(2 * kh) * D + col0 + m;    // W[k + 2*kh + {0,1}, col]

    v8f c = {};
#pragma unroll
    for (int k = 0; k < D; k += 4) {
        v2f a = { arow[k], arow[k + 1] };
        v2f b = { bcol[(size_t)k * D], bcol[(size_t)(k + 1) * D] };
        c = __builtin_amdgcn_wmma_f32_16x16x4_f32(
                /*neg_a=*/false, a, /*neg_b=*/false, b,
                /*c_mod=*/(short)0, c, /*reuse_a=*/false, /*reuse_b=*/false);
    }

    int col = col0 + m;
    float bb = bias[col];
#pragma unroll
    for (int j = 0; j < 8; ++j) {
        int r = row0 + j + 8 * kh;  // D layout: vgpr j -> row j + 8*(lane>>4)
        if (r < N) {
            float z = fmaxf(c[j] + bb, 0.f);
            size_t idx = (size_t)r * D + col;
            hout[idx] = hprev[idx] + z;
        }
    }
}

// ---------------- global attention pooling ----------------

__global__ void k_gate(const float* __restrict__ h2, const float* __restrict__ gw,
                       const float* __restrict__ gb, const int* __restrict__ gid,
                       float* __restrict__ gate, unsigned* __restrict__ gmax, int N) {
    int wid  = (int)(((long)blockIdx.x * blockDim.x + threadIdx.x) >> 5);
    int lane = threadIdx.x & 31;
    if (wid >= N) return;
    float4 hv = *(const float4*)(h2 + (size_t)wid * D + lane * 4);
    float4 wv = *(const float4*)(gw + lane * 4);
    float p = hv.x * wv.x + hv.y * wv.y + hv.z * wv.z + hv.w * wv.w;
    p += __shfl_xor(p, 16);
    p += __shfl_xor(p, 8);
    p += __shfl_xor(p, 4);
    p += __shfl_xor(p, 2);
    p += __shfl_xor(p, 1);
    if (lane == 0) {
        float g = p + gb[0];
        gate[wid] = g;
        unsigned u = __float_as_uint(g);
        u = (u & 0x80000000u) ? ~u : (u | 0x80000000u);  // order-preserving map
        atomicMax(gmax + gid[wid], u);
    }
}

__global__ void k_exp(float* __restrict__ gate, const unsigned* __restrict__ gmax,
                      const int* __restrict__ gid, float* __restrict__ denom, int N) {
    int i = blockIdx.x * blockDim.x + threadIdx.x;
    if (i >= N) return;
    int g = gid[i];
    unsigned u = gmax[g];
    unsigned fb = (u & 0x80000000u) ? (u ^ 0x80000000u) : ~u;
    float e = expf(gate[i] - __uint_as_float(fb));
    gate[i] = e;
    atomicAdd(denom + g, e);
}

__global__ void k_pool(const float* __restrict__ h2, const float* __restrict__ e,
                       const float* __restrict__ denom, const int* __restrict__ gid,
                       float* __restrict__ out, int N) {
    int wid  = (int)(((long)blockIdx.x * blockDim.x + threadIdx.x) >> 5);
    int lane = threadIdx.x & 31;
    if (wid >= N) return;
    int g = gid[wid];
    float coef = e[wid] / denom[g];
    float4 hv = *(const float4*)(h2 + (size_t)wid * D + lane * 4);
    float* dst = out + (size_t)g * D + lane * 4;
    atomicAdd(dst + 0, coef * hv.x); atomicAdd(dst + 1, coef * hv.y);
    atomicAdd(dst + 2, coef * hv.z); atomicAdd(dst + 3, coef * hv.w);
}

// ---------------- launch ----------------

static inline unsigned nblk(long n, int t) { return (unsigned)((n + t - 1) / t); }

extern "C" void kernel_launch(void* const* d_in, const int* in_sizes, int n_in,
                              void* d_out, int out_size, void* d_ws, size_t ws_size,
                              hipStream_t stream) {
    const float* h     = (const float*)d_in[0];
    const int*   src   = (const int*)d_in[1];
    const int*   dst   = (const int*)d_in[2];
    const int*   gid   = (const int*)d_in[3];
    const float* W1    = (const float*)d_in[5];
    const float* b1    = (const float*)d_in[6];
    const float* W2    = (const float*)d_in[7];
    const float* b2    = (const float*)d_in[8];
    const float* gW    = (const float*)d_in[9];
    const float* gB    = (const float*)d_in[10];
    const float* gamma = (const float*)d_in[11];
    const float* beta  = (const float*)d_in[12];

    int N = in_sizes[0] / D;
    int E = in_sizes[1];
    int G = out_size / (2 * D);

    float* out     = (float*)d_out;          // hg [G*D]
    float* out_avg = out + (size_t)G * D;    // init_avg_h [G*D]

    size_t ND = (size_t)N * D;
    float* buf0   = (float*)d_ws;            // xs, then h2
    float* buf1   = buf0 + ND;               // agg
    float* buf2   = buf1 + ND;               // h1
    float* invout = buf2 + ND;
    float* invin  = invout + N;
    float* gsum   = invin + N;
    float* gcnt   = gsum + (size_t)G * D;
    float* bsum   = gcnt + G;                // [D]
    float* bsq    = bsum + D;                // [D] (contiguous with bsum)
    float* bscale = bsq + D;
    float* bshift = bscale + D;
    float* gate   = bshift + D;              // [N], becomes e in-place
    unsigned* gmax = (unsigned*)(gate + N);  // [G]
    float* denom  = (float*)(gmax + G);      // [G]

    // init accumulators + output hg region
    k_zero<<<nblk((long)G * D, 256), 256, 0, stream>>>(gsum, (long)G * D);
    k_zero<<<nblk(G, 256), 256, 0, stream>>>(gcnt, G);
    k_zero<<<nblk(N, 256), 256, 0, stream>>>(invout, N);
    k_zero<<<nblk(N, 256), 256, 0, stream>>>(invin, N);
    k_zero<<<nblk((long)G * D, 256), 256, 0, stream>>>(out, (long)G * D);
    k_zero<<<nblk(G, 256), 256, 0, stream>>>(denom, G);
    k_init_gmax<<<nblk(G, 256), 256, 0, stream>>>(gmax, G);

    // graph means + degrees
    k_graph_stats<<<nblk((long)N * 32, 256), 256, 0, stream>>>(h, gid, gsum, gcnt, N);
    k_degrees<<<nblk(E, 256), 256, 0, stream>>>(src, dst, invout, invin, E);
    k_deg_finalize<<<nblk(N, 256), 256, 0, stream>>>(invout, invin, N);
    k_init_avg<<<nblk((long)G * D, 256), 256, 0, stream>>>(gsum, gcnt, out_avg, G * D);

    // two GCN layers
    const float* hin = h;
    const float* Ws[2]    = {W1, W2};
    const float* bsv[2]   = {b1, b2};
    float*       houts[2] = {buf2, buf0};    // h1 -> buf2, h2 -> buf0 (xs dead by then)
    for (int layer = 0; layer < 2; ++layer) {
        k_zero<<<nblk(2 * D, 256), 256, 0, stream>>>(bsum, 2 * D);
        k_bn_stats<<<512, 128, 0, stream>>>(hin, bsum, bsq, N);
        k_bn_finalize<<<1, 128, 0, stream>>>(bsum, bsq, gamma, beta, bscale, bshift, N);
        k_xs<<<nblk((long)N * (D / 4), 256), 256, 0, stream>>>(hin, bscale, bshift, invout, buf0, N);
        k_zero<<<nblk((long)ND, 256), 256, 0, stream>>>(buf1, (long)ND);
        k_spmm<<<nblk((long)E * 32, 256), 256, 0, stream>>>(src, dst, buf0, buf1, E);
        k_gemm_relu_res<<<nblk(((long)N + 15) / 16 * 1, 1), 256, 0, stream>>>(
            buf1, invin, Ws[layer], bsv[layer], hin, houts[layer], N);
        hin = houts[layer];
    }

    // global attention pooling over h2 (= buf0)
    float* h2 = buf0;
    k_gate<<<nblk((long)N * 32, 256), 256, 0, stream>>>(h2, gW, gB, gid, gate, gmax, N);
    k_exp<<<nblk(N, 256), 256, 0, stream>>>(gate, gmax, gid, denom, N);
    k_pool<<<nblk((long)N * 32, 256), 256, 0, stream>>>(h2, gate, denom, gid, out, N);
}